// GaussianRasterizer_90357521973622
// MI455X (gfx1250) — compile-verified
//
#include <hip/hip_runtime.h>
#include <hip/hip_bf16.h>

#define HH 128
#define WW 128
#define NG 1024
#define RSTRIDE 16  // floats per record (64 bytes)

typedef __attribute__((ext_vector_type(2))) float v2f;
typedef __attribute__((ext_vector_type(8))) float v8f;

// ---------------------------------------------------------------------------
// Kernel 1: project 1024 gaussians -> 16-float records + sort keys (-depth)
// record: [k0..k5, 0, 0, r, g, b, opac_eff, depth, 0, 0, 0]
// mahal(p,g) = k0*x^2 + k1*x*y + k2*y^2 + k3*x + k4*y + k5
// The two zero slots at 6,7 feed the hi-lane half of the second WMMA A-chunk,
// so the rasterizer needs no per-lane value selection.
// ---------------------------------------------------------------------------
__global__ __launch_bounds__(256) void gs_project(
    const float* __restrict__ pos, const float* __restrict__ scl,
    const float* __restrict__ rot, const float* __restrict__ col,
    const float* __restrict__ opa, const float* __restrict__ Km,
    float* __restrict__ rec, float* __restrict__ key) {
  int i = blockIdx.x * blockDim.x + threadIdx.x;
  if (i >= NG) return;
  float fx = Km[0], cx = Km[2], fy = Km[4], cy = Km[5];
  float X = pos[i * 3 + 0], Y = pos[i * 3 + 1], Z = pos[i * 3 + 2];
  float zi = 1.0f / (Z + 1e-6f);
  float mx = fx * X * zi + cx;
  float my = fy * Y * zi + cy;

  // quaternion -> rotation matrix
  float qw = rot[i * 4 + 0], qx = rot[i * 4 + 1], qy = rot[i * 4 + 2], qz = rot[i * 4 + 3];
  float qn = rsqrtf(qw * qw + qx * qx + qy * qy + qz * qz);
  qw *= qn; qx *= qn; qy *= qn; qz *= qn;
  float R[3][3];
  R[0][0] = 1.f - 2.f * (qy * qy + qz * qz);
  R[0][1] = 2.f * (qx * qy - qw * qz);
  R[0][2] = 2.f * (qx * qz + qw * qy);
  R[1][0] = 2.f * (qx * qy + qw * qz);
  R[1][1] = 1.f - 2.f * (qx * qx + qz * qz);
  R[1][2] = 2.f * (qy * qz - qw * qx);
  R[2][0] = 2.f * (qx * qz - qw * qy);
  R[2][1] = 2.f * (qy * qz + qw * qx);
  R[2][2] = 1.f - 2.f * (qx * qx + qy * qy);

  float s0 = scl[i * 3 + 0], s1 = scl[i * 3 + 1], s2 = scl[i * 3 + 2];
  float sq[3] = {s0 * s0, s1 * s1, s2 * s2};
  // cov3d = (R*S)(R*S)^T
  float M[3][3];
#pragma unroll
  for (int r = 0; r < 3; r++)
#pragma unroll
    for (int c = 0; c < 3; c++)
      M[r][c] = R[r][0] * R[c][0] * sq[0] + R[r][1] * R[c][1] * sq[1] + R[r][2] * R[c][2] * sq[2];

  float j00 = fx * zi, j11 = fy * zi;
  float j02 = -fx * X * zi * zi, j12 = -fy * Y * zi * zi;
  float JC0[3], JC1[3];
#pragma unroll
  for (int c = 0; c < 3; c++) {
    JC0[c] = j00 * M[0][c] + j02 * M[2][c];
    JC1[c] = j11 * M[1][c] + j12 * M[2][c];
  }
  float a  = JC0[0] * j00 + JC0[2] * j02 + 1e-4f;
  float b  = JC0[1] * j11 + JC0[2] * j12;
  float c_ = JC1[0] * j00 + JC1[2] * j02;
  float d  = JC1[1] * j11 + JC1[2] * j12 + 1e-4f;
  float inv = 1.0f / (a * d - b * c_);
  float ia = d * inv, ibc = -(b + c_) * inv, id = a * inv;

  bool inb = (mx >= -50.f) && (mx <= (float)WW + 50.f) && (my >= -50.f) && (my <= (float)HH + 50.f);
  float op = inb ? opa[i] : 0.f;

  float* r = rec + i * RSTRIDE;
  r[0] = ia; r[1] = ibc; r[2] = id;
  r[3] = -(2.f * ia * mx + ibc * my);
  r[4] = -(2.f * id * my + ibc * mx);
  r[5] = ia * mx * mx + ibc * mx * my + id * my * my;
  r[6] = 0.f; r[7] = 0.f;  // zero pad read by hi-lane A-chunk 1
  r[8] = col[i * 3 + 0]; r[9] = col[i * 3 + 1]; r[10] = col[i * 3 + 2];
  r[11] = op; r[12] = Z;
  r[13] = 0.f; r[14] = 0.f; r[15] = 0.f;
  key[i] = -Z;  // ascending sort on -depth == back-to-front
}

// ---------------------------------------------------------------------------
// Kernel 2: single-block bitonic sort of 1024 (key, idx) + record gather
// ---------------------------------------------------------------------------
__global__ __launch_bounds__(1024) void gs_sort(
    const float* __restrict__ key_in, const float* __restrict__ rec_in,
    float* __restrict__ rec_out) {
  __shared__ float k[NG];
  __shared__ int ix[NG];
  int t = threadIdx.x;
  k[t] = key_in[t];
  ix[t] = t;
  __syncthreads();
  for (int kk = 2; kk <= NG; kk <<= 1) {
    for (int j = kk >> 1; j > 0; j >>= 1) {
      int p = t ^ j;
      if (p > t) {
        bool up = ((t & kk) == 0);
        float ka = k[t], kb = k[p];
        if ((ka > kb) == up) {
          k[t] = kb; k[p] = ka;
          int tmp = ix[t]; ix[t] = ix[p]; ix[p] = tmp;
        }
      }
      __syncthreads();
    }
  }
  int src = ix[t];
#pragma unroll
  for (int m = 0; m < RSTRIDE; m += 4) {
    float4 v = *(const float4*)(rec_in + src * RSTRIDE + m);
    *(float4*)(rec_out + t * RSTRIDE + m) = v;
  }
}

// ---------------------------------------------------------------------------
// Kernel 3: WMMA rasterizer.
// Block = 256 threads (8 waves). Wave owns 16 pixels; loops 64 tiles of 16
// sorted gaussians. mahal via 2x V_WMMA_F32_16X16X4_F32 (rank-6 bilinear
// form, K padded to 8). D layout: M=gaussian, N=pixel; lane L holds pixel
// L%16 and gaussians {0..7} (L<16) or {8..15} (L>=16) of the tile.
// ---------------------------------------------------------------------------
__global__ __launch_bounds__(256) void gs_raster(
    const float* __restrict__ rec, const float* __restrict__ bg,
    float* __restrict__ out) {
  __shared__ float recs[NG * RSTRIDE];  // 64 KB of sorted gaussian records
  int tid = threadIdx.x;

  // Stage all records into LDS once (CDNA5 async-LDS path when available).
#if defined(__gfx1250__) && defined(__has_builtin)
#if __has_builtin(__builtin_amdgcn_global_load_async_to_lds_b128) && \
    __has_builtin(__builtin_amdgcn_s_wait_asynccnt)
#define GS_ASYNC_LDS 1
#endif
#endif
#ifdef GS_ASYNC_LDS
  {
    // Builtin signature (probe-confirmed via diagnostic):
    // (int4 addrspace(1)*, int4 addrspace(3)*, imm offset, imm cpol)
    typedef __attribute__((__vector_size__(4 * sizeof(int)))) int v4i;
    typedef __attribute__((address_space(1))) v4i* gptr_t;
    typedef __attribute__((address_space(3))) v4i* lptr_t;
    for (int i = tid; i < (NG * RSTRIDE) / 4; i += 256) {
      __builtin_amdgcn_global_load_async_to_lds_b128(
          (gptr_t)(rec + i * 4), (lptr_t)&recs[i * 4], 0, 0);
    }
    __builtin_amdgcn_s_wait_asynccnt(0);
  }
  __syncthreads();
#else
  for (int i = tid; i < (NG * RSTRIDE) / 4; i += 256)
    ((float4*)recs)[i] = ((const float4*)rec)[i];
  __syncthreads();
#endif

  int lane = tid & 31;
  int wave = tid >> 5;
  bool hi = lane >= 16;
  int p = blockIdx.x * 128 + wave * 16 + (lane & 15);
  float X = (float)(p & (WW - 1));
  float Y = (float)(p >> 7);

  // B matrix (4x16, K x pixel): features [x^2, xy, y^2, x] and [y, 1, 0, 0].
  // Layout: VGPR0 lanes0-15 = K0, lanes16-31 = K2; VGPR1 = K1 / K3.
  v2f b0, b1;
  b0.x = hi ? Y * Y : X * X;
  b0.y = hi ? X : X * Y;
  b1.x = hi ? 0.f : Y;
  b1.y = hi ? 0.f : 1.f;

  // Lane-constant LDS bases: A-chunk loads need no value selection because
  // the record layout supplies the hi-lane zeros directly.
  const float* laneRec = &recs[(lane & 15) * RSTRIDE];
  int offA0 = hi ? 2 : 0;  // k0,k1 | k2,k3
  int offA1 = hi ? 6 : 4;  // k4,k5 | 0,0
  const float* segRec = &recs[(hi ? 8 : 0) * RSTRIDE];

  float Gr = bg[0], Gg = bg[1], Gb = bg[2], Gd = 0.f, GT = 1.f;

#pragma unroll 2
  for (int t = 0; t < NG / 16; t++) {
    int tileOff = t * 16 * RSTRIDE;
    const float* rA = laneRec + tileOff;
    v2f a0, a1;
    a0.x = rA[offA0]; a0.y = rA[offA0 + 1];
    a1.x = rA[offA1]; a1.y = rA[offA1 + 1];

    v8f acc = {};
    acc = __builtin_amdgcn_wmma_f32_16x16x4_f32(false, a0, false, b0,
                                                (short)0, acc, false, false);
    acc = __builtin_amdgcn_wmma_f32_16x16x4_f32(false, a1, false, b1,
                                                (short)0, acc, false, false);

    // In-order composite of this lane's 8-gaussian segment.
    const float* rs = segRec + tileOff;
    float sr = 0.f, sg = 0.f, sb = 0.f, sd = 0.f, sT = 1.f;
#pragma unroll
    for (int r = 0; r < 8; r++) {
      float cr = rs[r * RSTRIDE + 8];
      float cg = rs[r * RSTRIDE + 9];
      float cb = rs[r * RSTRIDE + 10];
      float op = rs[r * RSTRIDE + 11];
      float dp = rs[r * RSTRIDE + 12];
      float w = __expf(-0.5f * acc[r]) * op;
      float alpha = (w > 0.01f) ? w : 0.f;
      float cA = sT * alpha;
      sr += cA * cr; sg += cA * cg; sb += cA * cb; sd += cA * dp;
      sT *= (1.f - alpha);
    }

    // Merge the two ordered segments per pixel (low lane = earlier gaussians).
    float or_ = __shfl_xor(sr, 16, 32);
    float og  = __shfl_xor(sg, 16, 32);
    float ob  = __shfl_xor(sb, 16, 32);
    float od  = __shfl_xor(sd, 16, 32);
    float oT  = __shfl_xor(sT, 16, 32);
    float f1r = hi ? or_ : sr, f1g = hi ? og : sg, f1b = hi ? ob : sb;
    float f1d = hi ? od : sd, f1T = hi ? oT : sT;
    float f2r = hi ? sr : or_, f2g = hi ? sg : og, f2b = hi ? sb : ob;
    float f2d = hi ? sd : od, f2T = hi ? sT : oT;
    float tr = f1r + f1T * f2r;
    float tg = f1g + f1T * f2g;
    float tb = f1b + f1T * f2b;
    float td = f1d + f1T * f2d;
    float tT = f1T * f2T;

    Gr += GT * tr; Gg += GT * tg; Gb += GT * tb; Gd += GT * td;
    GT *= tT;
  }

  if (!hi) {
    out[p * 3 + 0] = fminf(fmaxf(Gr, 0.f), 1.f);
    out[p * 3 + 1] = fminf(fmaxf(Gg, 0.f), 1.f);
    out[p * 3 + 2] = fminf(fmaxf(Gb, 0.f), 1.f);
    out[HH * WW * 3 + p] = Gd;
    out[HH * WW * 3 + HH * WW + p] = 1.f - GT;
  }
}

// ---------------------------------------------------------------------------
extern "C" void kernel_launch(void* const* d_in, const int* in_sizes, int n_in,
                              void* d_out, int out_size, void* d_ws, size_t ws_size,
                              hipStream_t stream) {
  const float* pos = (const float*)d_in[0];
  const float* scl = (const float*)d_in[1];
  const float* rot = (const float*)d_in[2];
  const float* col = (const float*)d_in[3];
  const float* opa = (const float*)d_in[4];
  const float* Km  = (const float*)d_in[5];
  const float* bg  = (const float*)d_in[6];
  float* out = (float*)d_out;
  float* ws = (float*)d_ws;

  float* rec_unsorted = ws;                       // 16384 floats
  float* keys         = ws + NG * RSTRIDE;        // 1024 floats
  float* rec_sorted   = ws + NG * RSTRIDE + NG;   // 16384 floats

  gs_project<<<NG / 256, 256, 0, stream>>>(pos, scl, rot, col, opa, Km,
                                           rec_unsorted, keys);
  gs_sort<<<1, NG, 0, stream>>>(keys, rec_unsorted, rec_sorted);
  gs_raster<<<(HH * WW) / 128, 256, 0, stream>>>(rec_sorted, bg, out);
}